// SumGNN_5875515261228
// MI455X (gfx1250) — compile-verified
//
#include <hip/hip_runtime.h>
#include <math.h>

typedef float v2f __attribute__((ext_vector_type(2)));
typedef float v8f __attribute__((ext_vector_type(8)));

#define NNODES 50000
#define NEDGES 800000
#define HID 128
#define NT 8                  // 128 cols / 16
#define WPB 8                 // waves per block (256 threads)
#define MTILES (NNODES / 16)  // 3125, exact
#define KSTEPS (HID / 4)      // 32 k-steps of 4 for f32 WMMA
#define WFRAGS (KSTEPS * NT * 32)  // 8192 v2f pairs = 64 KB

// ---------------------------------------------------------------------------
// Fused GEMM: out = epilogue(A1 @ W1 + b1 [+ A2 @ W2 + b2])
// One wave computes a 16 x 128 output strip with v_wmma_f32_16x16x4_f32.
// Weight matrix is staged once per workgroup into LDS in per-lane fragment
// order, so the hot loop does one contiguous ds_load_b64 per (k-step, n-tile)
// (bank-conflict-free) instead of per-wave redundant global loads.
// A: [M x 128] row-major.  W: [128 x 128] row-major (K x N).  b: [128].
// ---------------------------------------------------------------------------
__device__ __forceinline__ void stage_weights(const float* __restrict__ Wg,
                                              v2f* __restrict__ smem)
{
    // p = ((k4 * NT + j) * 32 + lane); fragment holds W[k][col], W[k+1][col]
    for (int p = threadIdx.x; p < WFRAGS; p += 256) {
        const int l   = p & 31;
        const int j   = (p >> 5) & 7;
        const int k4  = p >> 8;
        const int k   = k4 * 4 + (l >> 4) * 2;
        const int col = j * 16 + (l & 15);
        v2f w;
        w.x = Wg[k * HID + col];
        w.y = Wg[(k + 1) * HID + col];
        smem[p] = w;
    }
}

__global__ __launch_bounds__(256)
void gnn_gemm_fused(const float* __restrict__ A1,
                    const float* __restrict__ W1,
                    const float* __restrict__ b1,
                    const float* __restrict__ A2,
                    const float* __restrict__ W2,
                    const float* __restrict__ b2,
                    const float* resid,
                    float* __restrict__ out,
                    int do_gelu)
{
    __shared__ v2f smem[WFRAGS];          // 64 KB weight fragments

    const int lane  = threadIdx.x & 31;
    const int wave  = threadIdx.x >> 5;
    int tile        = blockIdx.x * WPB + wave;
    const bool active = (tile < MTILES);  // wave-uniform
    if (!active) tile = MTILES - 1;       // clamp for safe addressing

    const int m    = lane & 15;           // row / col within 16-tile
    const int half = lane >> 4;           // lane half selects K/M sub-slot
    const int arow = tile * 16 + m;       // A-fragment row for this lane

    v8f acc[NT];
#pragma unroll
    for (int j = 0; j < NT; ++j) acc[j] = (v8f)(0.0f);

    // ---- GEMM 1: A1 @ W1 -------------------------------------------------
    stage_weights(W1, smem);
    __syncthreads();
    {
        const float* ap = A1 + (size_t)arow * HID;
#pragma unroll 4
        for (int k4 = 0; k4 < KSTEPS; ++k4) {
            const v2f a = *(const v2f*)(ap + k4 * 4 + half * 2);
            const v2f* bp = smem + (k4 * NT) * 32 + lane;
#pragma unroll
            for (int j = 0; j < NT; ++j) {
                const v2f b = bp[j * 32];
                acc[j] = __builtin_amdgcn_wmma_f32_16x16x4_f32(
                    false, a, false, b, (short)0, acc[j], false, false);
            }
        }
    }

    // ---- GEMM 2 (neighbor aggregate path), uniform branch ----------------
    if (A2) {
        __syncthreads();
        stage_weights(W2, smem);
        __syncthreads();
        const float* ap = A2 + (size_t)arow * HID;
#pragma unroll 4
        for (int k4 = 0; k4 < KSTEPS; ++k4) {
            const v2f a = *(const v2f*)(ap + k4 * 4 + half * 2);
            const v2f* bp = smem + (k4 * NT) * 32 + lane;
#pragma unroll
            for (int j = 0; j < NT; ++j) {
                const v2f b = bp[j * 32];
                acc[j] = __builtin_amdgcn_wmma_f32_16x16x4_f32(
                    false, a, false, b, (short)0, acc[j], false, false);
            }
        }
    }

    // ---- Epilogue: bias, GELU(exact erf), residual, store ---------------
    if (active) {
#pragma unroll
        for (int j = 0; j < NT; ++j) {
            const int col = j * 16 + m;
            float bias = b1[col];
            if (b2) bias += b2[col];
#pragma unroll
            for (int r = 0; r < 8; ++r) {
                const int orow = tile * 16 + r + half * 8;  // C/D row layout
                float v = acc[j][r] + bias;
                if (do_gelu)
                    v = 0.5f * v * (1.0f + erff(v * 0.70710678118654752f));
                if (resid)
                    v += resid[(size_t)orow * HID + col];
                out[(size_t)orow * HID + col] = v;
            }
        }
    }
}

// ---------------------------------------------------------------------------
// Zero the aggregation buffer (float4-vectorized).
// ---------------------------------------------------------------------------
__global__ __launch_bounds__(256)
void gnn_zero(float4* __restrict__ p, int n4)
{
    int i = blockIdx.x * blockDim.x + threadIdx.x;
    if (i < n4) p[i] = make_float4(0.f, 0.f, 0.f, 0.f);
}

// ---------------------------------------------------------------------------
// Edge scatter-add: agg[dst] += h[src].  32 threads per edge, float4 gather,
// hardware fp32 atomics (global_atomic_add_f32). All node data is L2-resident
// (25.6 MB per tensor vs 192 MB L2), so this is L2-atomic-throughput bound.
// ---------------------------------------------------------------------------
__global__ __launch_bounds__(256)
void gnn_scatter(const float* __restrict__ h,
                 const int* __restrict__ src,
                 const int* __restrict__ dst,
                 float* __restrict__ agg)
{
    long long gid = (long long)blockIdx.x * blockDim.x + threadIdx.x;
    int e = (int)(gid >> 5);
    int c = ((int)gid & 31) * 4;
    if (e >= NEDGES) return;
    int s = src[e];
    int d = dst[e];
    const float4 v = *(const float4*)(h + (size_t)s * HID + c);
    float* base = agg + (size_t)d * HID + c;
    unsafeAtomicAdd(base + 0, v.x);
    unsafeAtomicAdd(base + 1, v.y);
    unsafeAtomicAdd(base + 2, v.z);
    unsafeAtomicAdd(base + 3, v.w);
}

// ---------------------------------------------------------------------------
extern "C" void kernel_launch(void* const* d_in, const int* in_sizes, int n_in,
                              void* d_out, int out_size, void* d_ws, size_t ws_size,
                              hipStream_t stream)
{
    (void)in_sizes; (void)n_in; (void)out_size; (void)ws_size;

    const float* x      = (const float*)d_in[0];
    const int*   edges  = (const int*)  d_in[1];   // [2, NEDGES]
    const float* enc_w  = (const float*)d_in[2];
    const float* enc_b  = (const float*)d_in[3];
    const float* self_w = (const float*)d_in[4];   // [3,128,128]
    const float* self_b = (const float*)d_in[5];   // [3,128]
    const float* nb_w   = (const float*)d_in[6];
    const float* nb_b   = (const float*)d_in[7];
    const float* dec_w  = (const float*)d_in[8];
    const float* dec_b  = (const float*)d_in[9];
    float* out = (float*)d_out;

    const size_t NH = (size_t)NNODES * HID;        // 6.4M floats = 25.6 MB
    float* h0  = (float*)d_ws;
    float* h1  = h0 + NH;
    float* agg = h1 + NH;                          // total 76.8 MB scratch

    const int* src = edges;
    const int* dst = edges + NEDGES;

    dim3 blk(256);
    const int gemm_grid    = (MTILES + WPB - 1) / WPB;               // 391
    const int zero_grid    = (int)((NH / 4 + 255) / 256);
    const int scatter_grid = (int)(((long long)NEDGES * 32 + 255) / 256);

    // Encoder: h0 = gelu(x @ enc_w + enc_b)
    gnn_gemm_fused<<<gemm_grid, blk, 0, stream>>>(
        x, enc_w, enc_b, nullptr, nullptr, nullptr, nullptr, h0, 1);

    float* hin = h0;
    float* hout = h1;
    for (int k = 0; k < 3; ++k) {
        gnn_zero<<<zero_grid, blk, 0, stream>>>((float4*)agg, (int)(NH / 4));
        gnn_scatter<<<scatter_grid, blk, 0, stream>>>(hin, src, dst, agg);
        // hout = gelu(hin@self_w[k]+b + agg@nb_w[k]+b) + hin
        gnn_gemm_fused<<<gemm_grid, blk, 0, stream>>>(
            hin, self_w + (size_t)k * HID * HID, self_b + (size_t)k * HID,
            agg, nb_w   + (size_t)k * HID * HID, nb_b   + (size_t)k * HID,
            hin, hout, 1);
        float* t = hin; hin = hout; hout = t;
    }

    // Decoder: out = h @ dec_w + dec_b   (no gelu, no residual)
    gnn_gemm_fused<<<gemm_grid, blk, 0, stream>>>(
        hin, dec_w, dec_b, nullptr, nullptr, nullptr, nullptr, out, 0);
}